// SelectiveMemory_57183194579414
// MI455X (gfx1250) — compile-verified
//
#include <hip/hip_runtime.h>
#include <math.h>

typedef __attribute__((ext_vector_type(2))) float v2f;
typedef __attribute__((ext_vector_type(8))) float v8f;

#define WMMA_F32(a, b, c) \
  __builtin_amdgcn_wmma_f32_16x16x4_f32(false, (a), false, (b), (short)0, (c), false, false)

// ---------------------------------------------------------------------------
// Q/K/V projections: O = X(Nx256) @ W(256x256) + bias for 3 weight sets.
// Block = 8 waves sharing one 16-wide W column block staged in LDS;
// each wave computes one 16x16 tile with V_WMMA_F32_16X16X4_F32.
// ---------------------------------------------------------------------------
__global__ void sm_qkv_gemm(const float* __restrict__ X,
                            const float* __restrict__ Wq, const float* __restrict__ bq,
                            const float* __restrict__ Wk, const float* __restrict__ bk,
                            const float* __restrict__ Wv, const float* __restrict__ bv,
                            float* __restrict__ Q, float* __restrict__ K,
                            float* __restrict__ V, int N) {
  __shared__ float sW[256 * 16];  // W column block, [k][n]
  const int colTiles = 48;        // 16 col tiles x 3 matrices
  int ct = blockIdx.x % colTiles;
  int rg = blockIdx.x / colTiles;
  int mat = ct >> 4;
  int col0 = (ct & 15) << 4;
  const float* W  = (mat == 0) ? Wq : (mat == 1) ? Wk : Wv;
  const float* Bb = (mat == 0) ? bq : (mat == 1) ? bk : bv;
  float* O        = (mat == 0) ? Q  : (mat == 1) ? K  : V;

  for (int idx = threadIdx.x; idx < 256 * 16; idx += blockDim.x)
    sW[idx] = W[(size_t)(idx >> 4) * 256 + col0 + (idx & 15)];
  __syncthreads();

  int wv = threadIdx.x >> 5;
  int lane = threadIdx.x & 31;
  int rowTile = rg * 8 + wv;
  if (rowTile * 16 >= N) return;  // wave-uniform: EXEC all-ones for WMMA
  int row0 = rowTile << 4;
  int mn = lane & 15;             // M for A-frag, N for B/C-frag
  int kh = lane >> 4;             // K-half (ISA 16x4 fp32 layout)
  v8f c = {0.f, 0.f, 0.f, 0.f, 0.f, 0.f, 0.f, 0.f};
  const float* xrow = X + (size_t)(row0 + mn) * 256 + 2 * kh;
  const float* swp = sW + 2 * kh * 16 + mn;
#pragma unroll 8
  for (int k = 0; k < 256; k += 4) {
    v2f a, b;
    a.x = xrow[k];        a.y = xrow[k + 1];
    b.x = swp[k * 16];    b.y = swp[k * 16 + 16];
    c = WMMA_F32(a, b, c);
  }
  float bias = Bb[col0 + mn];
#pragma unroll
  for (int r = 0; r < 8; ++r)
    O[(size_t)(row0 + r + 8 * kh) * 256 + col0 + mn] = c[r] + bias;
}

// ---------------------------------------------------------------------------
// Banded attention, fully on WMMA.  One wave handles 16 consecutive positions
// (never crossing a batch boundary: L and tile size are multiples of 16).
// Band columns j=0..79 map to pos = l0-64+j; row i attends j in [i+1, i+64];
// window16 -> d>=49, window4 -> d>=61 (suffixes share one softmax max).
// Out-of-batch rows use bk/bv (reference zero-pads hidden states).
// ---------------------------------------------------------------------------
#define PST 88  // padded LDS row stride (floats), 8B-aligned, low-conflict

__global__ void sm_attn_wmma(const float* __restrict__ Q, const float* __restrict__ K,
                             const float* __restrict__ V, const float* __restrict__ bk,
                             const float* __restrict__ bv, float* __restrict__ Mrg,
                             int N, int L) {
  __shared__ float sP[4 * 16 * PST];  // per-wave 16x80 score/prob tile
  int wv = threadIdx.x >> 5;
  int lane = threadIdx.x & 31;
  float* S = sP + wv * 16 * PST;
  int rt = blockIdx.x * 4 + wv;
  if (rt * 16 >= N) return;
  int l0g = rt << 4;      // global first row of tile
  int l0 = l0g % L;       // within-batch position
  int base = l0g - l0;    // batch base row
  int mn = lane & 15, kh = lane >> 4;

  // ---- scores: S(16x80) = Q_tile(16x256) @ K_band^T, scaled
  const float* qrow = Q + (size_t)(l0g + mn) * 256 + 2 * kh;
  for (int t = 0; t < 5; ++t) {
    int pos = l0 - 64 + t * 16 + mn;  // band column owned by this lane
    const float* krow =
        ((pos >= 0) ? (K + (size_t)(base + pos) * 256) : bk) + 2 * kh;
    v8f c = {0.f, 0.f, 0.f, 0.f, 0.f, 0.f, 0.f, 0.f};
#pragma unroll 8
    for (int k = 0; k < 256; k += 4) {
      v2f a, b;
      a.x = qrow[k];  a.y = qrow[k + 1];
      b.x = krow[k];  b.y = krow[k + 1];
      c = WMMA_F32(a, b, c);
    }
#pragma unroll
    for (int r = 0; r < 8; ++r)
      S[(r + 8 * kh) * PST + t * 16 + mn] = c[r] * 0.0625f;
  }

  // ---- softmax over 3 nested windows; lanes 0..15 own one row each.
  // All three softmaxes share the band max (weights are ratios).
  if (lane < 16) {
    int i = lane;
    float* row = S + i * PST;
    float m = -3.4e38f;
    for (int d = 1; d <= 64; ++d) m = fmaxf(m, row[i + d]);
    float s64 = 0.f, s16 = 0.f, s4 = 0.f;
    for (int d = 1; d <= 64; ++d) {
      float e = expf(row[i + d] - m);
      row[i + d] = e;
      s64 += e;
      s16 += (d >= 49) ? e : 0.f;
      s4  += (d >= 61) ? e : 0.f;
    }
    const float inv3 = 1.f / 3.f;
    float r64 = inv3 / s64, r16 = inv3 / s16, r4 = inv3 / s4;
    for (int j = 0; j < 80; ++j) {
      int d = j - i;
      float w = 0.f;
      if (d >= 1 && d <= 64) {
        float e = row[j];
        w = e * r64;
        if (d >= 49) w += e * r16;
        if (d >= 61) w += e * r4;
      }
      row[j] = w;  // combined mean-of-3-softmax weight
    }
  }
  // same-wave LDS ops are in-order; no cross-wave sharing of this slice.

  // ---- preload P A-fragments (reused across all 16 output col tiles)
  float areg[40];
#pragma unroll
  for (int s = 0; s < 20; ++s) {
    areg[2 * s]     = S[mn * PST + 4 * s + 2 * kh];
    areg[2 * s + 1] = S[mn * PST + 4 * s + 2 * kh + 1];
  }

  // ---- context: (16x256) = P(16x80) @ V_band(80x256)
  for (int t = 0; t < 16; ++t) {
    int col0 = t << 4;
    v8f c = {0.f, 0.f, 0.f, 0.f, 0.f, 0.f, 0.f, 0.f};
#pragma unroll
    for (int s = 0; s < 20; ++s) {
      int kr = 4 * s + 2 * kh;
      int p0 = l0 - 64 + kr;
      const float* v0 = (p0     >= 0) ? (V + (size_t)(base + p0)     * 256) : bv;
      const float* v1 = (p0 + 1 >= 0) ? (V + (size_t)(base + p0 + 1) * 256) : bv;
      v2f a, b;
      a.x = areg[2 * s];    a.y = areg[2 * s + 1];
      b.x = v0[col0 + mn];  b.y = v1[col0 + mn];
      c = WMMA_F32(a, b, c);
    }
#pragma unroll
    for (int r = 0; r < 8; ++r)
      Mrg[(size_t)(l0g + r + 8 * kh) * 256 + col0 + mn] = c[r];
  }
}

// ---------------------------------------------------------------------------
// Out = H + Mrg @ Wo + bo  (LDS-staged Wo column block, fused bias+residual)
// ---------------------------------------------------------------------------
__global__ void sm_out_gemm(const float* __restrict__ Mrg,
                            const float* __restrict__ Wo,
                            const float* __restrict__ bo,
                            const float* __restrict__ H,
                            float* __restrict__ Out, int N) {
  __shared__ float sW[256 * 16];
  const int colTiles = 16;
  int ct = blockIdx.x % colTiles;
  int rg = blockIdx.x / colTiles;
  int col0 = ct << 4;

  for (int idx = threadIdx.x; idx < 256 * 16; idx += blockDim.x)
    sW[idx] = Wo[(size_t)(idx >> 4) * 256 + col0 + (idx & 15)];
  __syncthreads();

  int wv = threadIdx.x >> 5;
  int lane = threadIdx.x & 31;
  int rowTile = rg * 8 + wv;
  if (rowTile * 16 >= N) return;
  int row0 = rowTile << 4;
  int mn = lane & 15, kh = lane >> 4;
  v8f c = {0.f, 0.f, 0.f, 0.f, 0.f, 0.f, 0.f, 0.f};
  const float* xrow = Mrg + (size_t)(row0 + mn) * 256 + 2 * kh;
  const float* swp = sW + 2 * kh * 16 + mn;
#pragma unroll 8
  for (int k = 0; k < 256; k += 4) {
    v2f a, b;
    a.x = xrow[k];      a.y = xrow[k + 1];
    b.x = swp[k * 16];  b.y = swp[k * 16 + 16];
    c = WMMA_F32(a, b, c);
  }
  float bias = bo[col0 + mn];
#pragma unroll
  for (int r = 0; r < 8; ++r) {
    size_t o = (size_t)(row0 + r + 8 * kh) * 256 + col0 + mn;
    Out[o] = c[r] + bias + H[o];
  }
}

extern "C" void kernel_launch(void* const* d_in, const int* in_sizes, int n_in,
                              void* d_out, int out_size, void* d_ws, size_t ws_size,
                              hipStream_t stream) {
  const float* H  = (const float*)d_in[0];
  const float* Wq = (const float*)d_in[1];
  const float* bq = (const float*)d_in[2];
  const float* Wk = (const float*)d_in[3];
  const float* bk = (const float*)d_in[4];
  const float* Wv = (const float*)d_in[5];
  const float* bv = (const float*)d_in[6];
  const float* Wo = (const float*)d_in[7];
  const float* bo = (const float*)d_in[8];
  float* Out = (float*)d_out;

  int N = in_sizes[0] / 256;  // B*L rows
  int L = 4096;               // per reference
  if (N < L) L = N;

  float* Q = (float*)d_ws;
  float* K = Q + (size_t)N * 256;
  float* V = K + (size_t)N * 256;
  float* M = V + (size_t)N * 256;

  int rowTiles = (N + 15) / 16;
  int rowGroups = (rowTiles + 7) / 8;

  sm_qkv_gemm<<<48 * rowGroups, 256, 0, stream>>>(H, Wq, bq, Wk, bk, Wv, bv,
                                                  Q, K, V, N);
  sm_attn_wmma<<<(rowTiles + 3) / 4, 128, 0, stream>>>(Q, K, V, bk, bv, M, N, L);
  sm_out_gemm<<<16 * rowGroups, 256, 0, stream>>>(M, Wo, bo, H, Out, N);
}